// L_Mask_43679817400497
// MI455X (gfx1250) — compile-verified
//
#include <hip/hip_runtime.h>

#define BATCH 8
#define HH 768
#define WW 768
#define HW (HH * WW)                              // 589824
#define NBINS 256
#define NIMG 16                                   // 2 sources x 8 batches; n = src*8 + b
#define PIX_PER_BLOCK 1024                        // 256 threads x 4 pixels
#define BLOCKS_PER_BATCH (HW / PIX_PER_BLOCK)     // 576
#define NBLK (BATCH * BLOCKS_PER_BATCH)           // 4608

typedef float v2f __attribute__((ext_vector_type(2)));
typedef float v8f __attribute__((ext_vector_type(8)));

struct alignas(16) f4 { float f[4]; };

__device__ __forceinline__ f4 ldg4(const float* p) { return *(const f4*)p; }

__device__ __forceinline__ int quant(float g) {
  int q = (int)rintf(g);            // RNE, matches jnp.round
  q = q < 0 ? 0 : q;
  q = q > (NBINS - 1) ? (NBINS - 1) : q;
  return q;
}

// ---------------- K0: zero the 16 histograms ----------------
__global__ void k_zero_hist(unsigned int* __restrict__ hist) {
  hist[blockIdx.x * NBINS + threadIdx.x] = 0u;
}

// ---------------- K1: gray + per-block LDS histogram ----------------
__global__ void k_gray_hist(const float* __restrict__ vis,
                            const float* __restrict__ ir,
                            unsigned int* __restrict__ hist) {
  __shared__ unsigned int lh[2 * NBINS];
  int t = threadIdx.x;
  lh[t] = 0u; lh[t + NBINS] = 0u;
  __syncthreads();

  int b = blockIdx.x / BLOCKS_PER_BATCH;
  size_t pix0 = (size_t)(blockIdx.x % BLOCKS_PER_BATCH) * PIX_PER_BLOCK + (size_t)t * 4;
  size_t base = (size_t)b * 3 * HW + pix0;

  f4 vr = ldg4(vis + base), vg = ldg4(vis + base + HW), vb = ldg4(vis + base + 2 * HW);
  f4 ur = ldg4(ir  + base), ug = ldg4(ir  + base + HW), ub = ldg4(ir  + base + 2 * HW);

#pragma unroll
  for (int c = 0; c < 4; ++c) {
    int qv = quant(0.299f * vr.f[c] + 0.587f * vg.f[c] + 0.114f * vb.f[c]);
    int qi = quant(0.299f * ur.f[c] + 0.587f * ug.f[c] + 0.114f * ub.f[c]);
    atomicAdd(&lh[qv], 1u);
    atomicAdd(&lh[NBINS + qi], 1u);
  }
  __syncthreads();
  atomicAdd(&hist[(0 * BATCH + b) * NBINS + t], lh[t]);
  atomicAdd(&hist[(1 * BATCH + b) * NBINS + t], lh[t + NBINS]);
}

// ---------------- K2: sal[n][i] = sum_k |i-k| * hist[n][k] via WMMA ----------------
// C[256,16] = D[256,256] @ H[256,16]; one wave per 16-row tile of D.
__global__ void k_sal_wmma(const unsigned int* __restrict__ hist,
                           float* __restrict__ sal) {
  __shared__ float hf[NBINS * NIMG];   // hf[k*16 + n] = (float)hist[n][k]
  int lane = threadIdx.x;              // 0..31, one wave
  for (int i = lane; i < NBINS * NIMG; i += 32) {
    int k = i >> 4, n = i & 15;
    hf[i] = (float)hist[n * NBINS + k];
  }
  __syncthreads();

  int m0  = blockIdx.x * 16;
  int row = lane & 15;                 // M (for A) / N (for B) index of this lane
  int kh  = (lane >> 4) * 2;           // lanes 0-15 -> K0/K1, lanes 16-31 -> K2/K3
  v8f c = {0.f, 0.f, 0.f, 0.f, 0.f, 0.f, 0.f, 0.f};

#if __has_builtin(__builtin_amdgcn_wmma_f32_16x16x4_f32)
#pragma unroll 8
  for (int k0 = 0; k0 < NBINS; k0 += 4) {
    v2f a, bb;
    a.x = fabsf((float)(m0 + row - (k0 + kh)));
    a.y = fabsf((float)(m0 + row - (k0 + kh + 1)));
    bb.x = hf[(k0 + kh) * 16 + row];
    bb.y = hf[(k0 + kh + 1) * 16 + row];
    c = __builtin_amdgcn_wmma_f32_16x16x4_f32(false, a, false, bb,
                                              (short)0, c, false, false);
  }
  // C/D layout: lanes 0-15 hold M = v, lanes 16-31 hold M = v + 8
  int n  = lane & 15;
  int rb = m0 + ((lane >> 4) * 8);
#pragma unroll
  for (int v = 0; v < 8; ++v)
    sal[n * NBINS + rb + v] = c[v];
#else
  // scalar fallback (not expected on gfx1250)
  for (int i = lane; i < 16 * NIMG; i += 32) {
    int r = m0 + (i >> 4);
    int n = i & 15;
    float s = 0.f;
    for (int k = 0; k < NBINS; ++k)
      s += fabsf((float)(r - k)) * hf[k * 16 + n];
    sal[n * NBINS + r] = s;
  }
#endif
}

// ---------------- K3: mx = max over occupied bins; normalize sal in place ----------------
__global__ void k_normalize(const unsigned int* __restrict__ hist,
                            float* __restrict__ sal,
                            float* __restrict__ invmx) {
  __shared__ float red[NBINS];
  __shared__ float inv;
  int n = blockIdx.x, t = threadIdx.x;
  float s = sal[n * NBINS + t];
  red[t] = (hist[n * NBINS + t] > 0u) ? s : 0.0f;   // smap max == max over occupied bins
  __syncthreads();
  for (int st = NBINS / 2; st > 0; st >>= 1) {
    if (t < st) red[t] = fmaxf(red[t], red[t + st]);
    __syncthreads();
  }
  if (t == 0) {
    float mx = red[0];
    inv = (mx > 0.0f) ? (1.0f / mx) : 0.0f;         // where(mx==0, 0, smap/mx)
    invmx[n] = inv;
  }
  __syncthreads();
  sal[n * NBINS + t] = s * inv;                     // sal now holds normalized map per bin
}

// ---------------- K4: fused weighting + per-block L1 partial sums ----------------
__global__ void k_loss(const float* __restrict__ vis,
                       const float* __restrict__ ir,
                       const float* __restrict__ fus,
                       const float* __restrict__ nmap,
                       float* __restrict__ partials) {
  __shared__ float mv[NBINS];
  __shared__ float mi[NBINS];
  __shared__ float red[256];
  int t = threadIdx.x;
  int b = blockIdx.x / BLOCKS_PER_BATCH;
  mv[t] = nmap[(0 * BATCH + b) * NBINS + t];
  mi[t] = nmap[(1 * BATCH + b) * NBINS + t];
  __syncthreads();

  size_t pix0 = (size_t)(blockIdx.x % BLOCKS_PER_BATCH) * PIX_PER_BLOCK + (size_t)t * 4;
  size_t base = (size_t)b * 3 * HW + pix0;

  f4 vr = ldg4(vis + base), vg = ldg4(vis + base + HW), vb = ldg4(vis + base + 2 * HW);
  f4 ur = ldg4(ir  + base), ug = ldg4(ir  + base + HW), ub = ldg4(ir  + base + 2 * HW);
  f4 fr = ldg4(fus + base), fg = ldg4(fus + base + HW), fb = ldg4(fus + base + 2 * HW);

  float acc = 0.f;
#pragma unroll
  for (int c = 0; c < 4; ++c) {
    int qv = quant(0.299f * vr.f[c] + 0.587f * vg.f[c] + 0.114f * vb.f[c]);
    int qi = quant(0.299f * ur.f[c] + 0.587f * ug.f[c] + 0.114f * ub.f[c]);
    float w1 = 0.4f + mi[qi] - 0.4f * mv[qv];
    float w2 = 1.0f - w1;
    acc += fabsf(w1 * vr.f[c] + w2 * ur.f[c] - fr.f[c])
         + fabsf(w1 * vg.f[c] + w2 * ug.f[c] - fg.f[c])
         + fabsf(w1 * vb.f[c] + w2 * ub.f[c] - fb.f[c]);
  }
  red[t] = acc;
  __syncthreads();
  for (int st = 128; st > 0; st >>= 1) {
    if (t < st) red[t] += red[t + st];
    __syncthreads();
  }
  if (t == 0) partials[blockIdx.x] = red[0];
}

// ---------------- K5: deterministic final reduction (f64) ----------------
__global__ void k_reduce(const float* __restrict__ partials, float* __restrict__ out) {
  __shared__ double red[256];
  int t = threadIdx.x;
  double s = 0.0;
  for (int i = t; i < NBLK; i += 256) s += (double)partials[i];
  red[t] = s;
  __syncthreads();
  for (int st = 128; st > 0; st >>= 1) {
    if (t < st) red[t] += red[t + st];
    __syncthreads();
  }
  if (t == 0) out[0] = (float)(red[0] / (double)((size_t)BATCH * 3 * HW));
}

extern "C" void kernel_launch(void* const* d_in, const int* in_sizes, int n_in,
                              void* d_out, int out_size, void* d_ws, size_t ws_size,
                              hipStream_t stream) {
  const float* vis = (const float*)d_in[0];
  const float* ir  = (const float*)d_in[1];
  const float* fus = (const float*)d_in[2];
  float* out = (float*)d_out;

  // workspace layout (all 4-byte elements):
  unsigned int* hist = (unsigned int*)d_ws;            // [16][256] u32   (16 KB)
  float* sal   = (float*)d_ws + NIMG * NBINS;          // [16][256] f32   (16 KB)
  float* invmx = sal + NIMG * NBINS;                   // [16] f32 (+pad)
  float* partials = invmx + 32;                        // [NBLK] f32      (18 KB)

  k_zero_hist<<<NIMG, NBINS, 0, stream>>>(hist);
  k_gray_hist<<<NBLK, 256, 0, stream>>>(vis, ir, hist);
  k_sal_wmma<<<NBINS / 16, 32, 0, stream>>>(hist, sal);
  k_normalize<<<NIMG, NBINS, 0, stream>>>(hist, sal, invmx);
  k_loss<<<NBLK, 256, 0, stream>>>(vis, ir, fus, sal, partials);
  k_reduce<<<1, 256, 0, stream>>>(partials, out);
}